// TLocVLBERT_19069654794252
// MI455X (gfx1250) — compile-verified
//
#include <hip/hip_runtime.h>
#include <hip/hip_bf16.h>

typedef __attribute__((ext_vector_type(2))) float v2f;
typedef __attribute__((ext_vector_type(8))) float v8f;

// ---------------------------------------------------------------------------
// Generic fp32 WMMA GEMM: C[M,N] = act(A[M,K](lda) @ W[K,N](ldw) + bias)
// One wave per 16x16 output tile; 4 waves (128 thr) per block along N.
// Native V_WMMA_F32_16X16X4_F32 keeps full fp32 accuracy (problem is far from
// the matrix-op roofline, so no reason to downconvert).
//
// Bounds handling: WMMA D[m][n] depends only on A row m and B col n, so OOB
// rows/cols are handled by CLAMPING the load addresses (no faults) and masking
// the store — the inner loop is completely branch-free, full-EXEC (as WMMA
// requires), with strength-reduced pointer increments only.
// ---------------------------------------------------------------------------
__global__ void gemm_wmma_f32(const float* __restrict__ A, int lda,
                              const float* __restrict__ W, int ldw,
                              const float* __restrict__ bias,
                              float* __restrict__ C, int ldc,
                              int M, int N, int K, int relu)
{
    const int wave  = threadIdx.x >> 5;
    const int lane  = threadIdx.x & 31;
    const int tileN = (blockIdx.x * 4 + wave) * 16;
    const int tileM = blockIdx.y * 16;
    if (tileN >= N) return;                 // uniform per wave

    const int l16   = lane & 15;
    const int khalf = lane >> 4;            // 0 or 1

    const int am  = tileM + l16;
    const int bn  = tileN + l16;
    const int amC = (am < M) ? am : (M - 1);   // clamped (garbage rows masked at store)
    const int bnC = (bn < N) ? bn : (N - 1);   // clamped (garbage cols masked at store)

    const float* __restrict__ aPtr = A + (size_t)amC * lda + 2 * khalf;
    const float* __restrict__ bPtr = W + (size_t)(2 * khalf) * ldw + bnC;
    const size_t bStep = 4 * (size_t)ldw;

    v8f acc = {};
#pragma unroll 4
    for (int k0 = 0; k0 < K; k0 += 4) {
        const v2f afrag = *(const v2f*)aPtr;          // A[m][ka], A[m][ka+1]
        v2f bfrag;
        bfrag.x = bPtr[0];                            // B[ka][n]
        bfrag.y = bPtr[ldw];                          // B[ka+1][n]
        acc = __builtin_amdgcn_wmma_f32_16x16x4_f32(
            /*neg_a=*/false, afrag, /*neg_b=*/false, bfrag,
            /*c_mod=*/(short)0, acc, /*reuse_a=*/false, /*reuse_b=*/false);
        aPtr += 4;
        bPtr += bStep;
    }

    const bool bnOK = bn < N;
    const float bv  = (bias != nullptr) ? bias[bnC] : 0.f;
#pragma unroll
    for (int r = 0; r < 8; ++r) {
        const int row = tileM + r + 8 * khalf;
        if (row < M && bnOK) {
            float v = acc[r] + bv;
            if (relu) v = fmaxf(v, 0.f);
            C[(size_t)row * ldc + bn] = v;
        }
    }
}

// ---------------------------------------------------------------------------
// Row-wise matvec epilogue for the visual heads: out[row*3] = Hh[row,:]·w2 + b2
// One wave per row, wave32 shuffle reduction.
// ---------------------------------------------------------------------------
__global__ void head_reduce(const float* __restrict__ Hh,
                            const float* __restrict__ w2,
                            const float* __restrict__ b2,
                            float* __restrict__ out, int M)
{
    const int row  = blockIdx.x * (blockDim.x >> 5) + (threadIdx.x >> 5);
    const int lane = threadIdx.x & 31;
    if (row >= M) return;
    const float* __restrict__ hr = Hh + (size_t)row * 512;
    float s = 0.f;
#pragma unroll 4
    for (int k = lane; k < 512; k += 32) s += hr[k] * w2[k];
#pragma unroll
    for (int off = 16; off > 0; off >>= 1) s += __shfl_down(s, off, 32);
    if (lane == 0) out[(size_t)row * 3] = s + b2[0];
}

// ---------------------------------------------------------------------------
// Fused IoU head: out[b,c,i,j] = sum_k relu(pS[b,i,k]+ps[b,(i+j)/2,k]+pE[b,j,k]
//                 + b3a[k]) * w3b[k,c] + b3b[c].
// One workgroup per (b,i); pS row / b3a / w3b staged in LDS; thread = j.
// Avoids ever materializing the 270 MB grid tensor.
// ---------------------------------------------------------------------------
__global__ void iou_fused(const float* __restrict__ pS,
                          const float* __restrict__ ps,
                          const float* __restrict__ pE,
                          const float* __restrict__ b3a,
                          const float* __restrict__ w3b,
                          const float* __restrict__ b3b,
                          float* __restrict__ out)   // (8,3,129,129)
{
    __shared__ float sPS[512];
    __shared__ float sB[512];
    __shared__ float sW[512 * 3];

    const int b = blockIdx.x / 129;
    const int i = blockIdx.x % 129;

    for (int k = threadIdx.x; k < 512; k += blockDim.x) {
        sPS[k] = pS[((size_t)(b * 129 + i)) * 512 + k];
        sB[k]  = b3a[k];
    }
    for (int k = threadIdx.x; k < 1536; k += blockDim.x) sW[k] = w3b[k];
    __syncthreads();

    const int j = threadIdx.x;
    if (j >= 129) return;
    const int idx = (i + j) >> 1;
    const float4* __restrict__ psr = (const float4*)(ps + ((size_t)(b * 129 + idx)) * 512);
    const float4* __restrict__ per = (const float4*)(pE + ((size_t)(b * 129 + j)) * 512);

    float a0 = 0.f, a1 = 0.f, a2 = 0.f;
    for (int k4 = 0; k4 < 128; ++k4) {
        const float4 x = psr[k4];
        const float4 y = per[k4];
        const int k = k4 * 4;
        float g;
        g = fmaxf(sPS[k + 0] + x.x + y.x + sB[k + 0], 0.f);
        a0 += g * sW[(k + 0) * 3 + 0]; a1 += g * sW[(k + 0) * 3 + 1]; a2 += g * sW[(k + 0) * 3 + 2];
        g = fmaxf(sPS[k + 1] + x.y + y.y + sB[k + 1], 0.f);
        a0 += g * sW[(k + 1) * 3 + 0]; a1 += g * sW[(k + 1) * 3 + 1]; a2 += g * sW[(k + 1) * 3 + 2];
        g = fmaxf(sPS[k + 2] + x.z + y.z + sB[k + 2], 0.f);
        a0 += g * sW[(k + 2) * 3 + 0]; a1 += g * sW[(k + 2) * 3 + 1]; a2 += g * sW[(k + 2) * 3 + 2];
        g = fmaxf(sPS[k + 3] + x.w + y.w + sB[k + 3], 0.f);
        a0 += g * sW[(k + 3) * 3 + 0]; a1 += g * sW[(k + 3) * 3 + 1]; a2 += g * sW[(k + 3) * 3 + 2];
    }
    const size_t TT = 129 * 129;
    const size_t base = ((size_t)b * 3) * TT + (size_t)i * 129 + (size_t)j;
    out[base]          = a0 + b3b[0];
    out[base + TT]     = a1 + b3b[1];
    out[base + 2 * TT] = a2 + b3b[2];
}

// ---------------------------------------------------------------------------
// IoU mask map: closed-form predicate on (i, d=j-i).
// ---------------------------------------------------------------------------
__global__ void mask_map(float* __restrict__ out)
{
    const int t = blockIdx.x * blockDim.x + threadIdx.x;
    if (t >= 129 * 129) return;
    const int i = t / 129, j = t % 129;
    const int d = j - i;
    float v = 0.f;
    if (i < 128) {
        if (d >= 1 && d <= 16) v = 1.f;
        if ((i % 2 == 0) && i <= 110 && d >= 18 && d <= 32 && (d % 2 == 0)) v = 1.f;
        if ((i % 4 == 0) && i <= 92  && d >= 36 && d <= 64 && (d % 4 == 0)) v = 1.f;
        if ((i % 8 == 0) && i <= 56  && d >= 72 && (d % 8 == 0)) v = 1.f;
    }
    out[t] = v;
}

// ---------------------------------------------------------------------------
// Host-side launch
// ---------------------------------------------------------------------------
static inline void launch_gemm(const float* A, int lda, const float* W, int ldw,
                               const float* bias, float* C, int ldc,
                               int M, int N, int K, int relu, hipStream_t s)
{
    dim3 grid((N + 63) / 64, (M + 15) / 16);
    gemm_wmma_f32<<<grid, 128, 0, s>>>(A, lda, W, ldw, bias, C, ldc, M, N, K, relu);
}

extern "C" void kernel_launch(void* const* d_in, const int* in_sizes, int n_in,
                              void* d_out, int out_size, void* d_ws, size_t ws_size,
                              hipStream_t stream)
{
    (void)in_sizes; (void)n_in; (void)out_size; (void)ws_size;

    const float* text = (const float*)d_in[0];   // (8,100,512)
    const float* obj  = (const float*)d_in[1];   // (8,129,512)
    const float* w_t1 = (const float*)d_in[2];
    const float* b_t1 = (const float*)d_in[3];
    const float* w_t2 = (const float*)d_in[4];
    const float* b_t2 = (const float*)d_in[5];
    const float* w2   = (const float*)d_in[6];
    const float* b2   = (const float*)d_in[7];
    const float* w3a  = (const float*)d_in[8];   // (1536,512)
    const float* b3a  = (const float*)d_in[9];
    const float* w3b  = (const float*)d_in[10];  // (512,3)
    const float* b3b  = (const float*)d_in[11];
    const float* w_s1 = (const float*)d_in[12];
    const float* b_s1 = (const float*)d_in[13];
    const float* w_s2 = (const float*)d_in[14];
    const float* b_s2 = (const float*)d_in[15];
    const float* w_e1 = (const float*)d_in[16];
    const float* b_e1 = (const float*)d_in[17];
    const float* w_e2 = (const float*)d_in[18];
    const float* b_e2 = (const float*)d_in[19];
    const float* w_c1 = (const float*)d_in[20];
    const float* b_c1 = (const float*)d_in[21];
    const float* w_c2 = (const float*)d_in[22];
    const float* b_c2 = (const float*)d_in[23];

    const int MT = 8 * 100;   // 800 text rows
    const int MO = 8 * 129;   // 1032 object rows

    // Output layout (flat, reference return order)
    float* out_text   = (float*)d_out;                       // 800*1000
    float* out_visual = out_text + 800 * 1000;               // 1032*3
    float* out_iou    = out_visual + MO * 3;                 // 8*3*129*129
    float* out_mask   = out_iou + 8 * 3 * 129 * 129;         // 129*129

    // Workspace layout (floats)
    float* ws  = (float*)d_ws;
    float* H1  = ws;                        // 800*512
    float* h   = H1 + 800 * 512;            // 1032*1536
    float* pS  = h + (size_t)MO * 1536;     // 1032*512
    float* psC = pS + (size_t)MO * 512;     // 1032*512
    float* pE  = psC + (size_t)MO * 512;    // 1032*512
    float* Hs  = pE + (size_t)MO * 512;     // 1032*512
    float* He  = Hs + (size_t)MO * 512;     // 1032*512
    float* Hc  = He + (size_t)MO * 512;     // 1032*512

    // 1) Text head
    launch_gemm(text, 512, w_t1, 512, b_t1, H1, 512, MT, 512, 512, 1, stream);
    launch_gemm(H1, 512, w_t2, 1000, b_t2, out_text, 1000, MT, 1000, 512, 0, stream);

    // 2) h = relu(obj @ w2 + b2)  -> [hs | hc | he]
    launch_gemm(obj, 512, w2, 1536, b2, h, 1536, MO, 1536, 512, 1, stream);

    // 3) pS = hs @ w3a[:512], ps = hc @ w3a[512:1024], pE = he @ w3a[1024:]
    launch_gemm(h,        1536, w3a,              512, nullptr, pS,  512, MO, 512, 512, 0, stream);
    launch_gemm(h + 512,  1536, w3a + 512 * 512,  512, nullptr, psC, 512, MO, 512, 512, 0, stream);
    launch_gemm(h + 1024, 1536, w3a + 1024 * 512, 512, nullptr, pE,  512, MO, 512, 512, 0, stream);

    // 4) Visual heads: hidden GEMMs + matvec reductions (cols 0:s/hs, 1:e/he, 2:c/hc)
    launch_gemm(h,        1536, w_s1, 512, b_s1, Hs, 512, MO, 512, 512, 1, stream);
    launch_gemm(h + 1024, 1536, w_e1, 512, b_e1, He, 512, MO, 512, 512, 1, stream);
    launch_gemm(h + 512,  1536, w_c1, 512, b_c1, Hc, 512, MO, 512, 512, 1, stream);

    const int rb = 256, rw = rb / 32;
    head_reduce<<<(MO + rw - 1) / rw, rb, 0, stream>>>(Hs, w_s2, b_s2, out_visual + 0, MO);
    head_reduce<<<(MO + rw - 1) / rw, rb, 0, stream>>>(He, w_e2, b_e2, out_visual + 1, MO);
    head_reduce<<<(MO + rw - 1) / rw, rb, 0, stream>>>(Hc, w_c2, b_c2, out_visual + 2, MO);

    // 5) Fused IoU grid + projection (never materializes (8,129,129,512))
    iou_fused<<<8 * 129, 160, 0, stream>>>(pS, psC, pE, b3a, w3b, b3b, out_iou);

    // 6) Constant mask map
    mask_map<<<(129 * 129 + 255) / 256, 256, 0, stream>>>(out_mask);
}